// HeteroInnerProduct_13846974562750
// MI455X (gfx1250) — compile-verified
//
#include <hip/hip_runtime.h>
#include <stdint.h>

typedef __attribute__((ext_vector_type(2))) float v2f;
typedef __attribute__((ext_vector_type(8))) float v8f;

#define WAVES_PER_BLOCK 8
#define THREADS (WAVES_PER_BLOCK * 32)
#define ROW_DW 36   /* 144-byte padded row stride in staging (16B aligned, bank-conflict-free frags) */
#define C_DW 12     /* 48-byte per-lane stride in C spill (16B aligned) */

// Each wave processes a tile of 16 edges:
//   rows 0..15  of staging = feat[src[e]]  (A matrix, M = edge)
//   rows 16..31 of staging = feat[dst[e]]  (B matrix, N = edge, K-major per row)
// 8x V_WMMA_F32_16X16X4_F32 accumulate C[m][n] = src_m . dst_n ; diagonal = output.
__global__ __launch_bounds__(THREADS) void hetero_dot_wmma_kernel(
    const float* __restrict__ feat,
    const int*   __restrict__ src,
    const int*   __restrict__ dst,
    float*       __restrict__ out,
    int nEdges, int nTiles)
{
    __shared__ float stage[WAVES_PER_BLOCK][32 * ROW_DW];
    __shared__ float cbuf [WAVES_PER_BLOCK][32 * C_DW];

    const int lane   = threadIdx.x & 31;
    const int wave   = threadIdx.x >> 5;
    const int gwave  = blockIdx.x * WAVES_PER_BLOCK + wave;
    const int nWaves = gridDim.x * WAVES_PER_BLOCK;

    float* st = stage[wave];
    float* cb = cbuf[wave];

    // Low 32 bits of the generic (flat) address of a __shared__ object are the
    // LDS byte offset (LDS aperture mapping discards addr[63:32]).
    const uint32_t stBase = (uint32_t)(uintptr_t)st;
    const uint64_t featBase = (uint64_t)(uintptr_t)feat;   // SGPR base for GVS async loads

    const int half  = lane >> 4;      // 0 = src side, 1 = dst side
    const int l16   = lane & 15;
    const int chunk = lane & 7;       // which 16B chunk of a 128B row
    const int rsel  = lane >> 3;      // 0..3: row sub-select per gather instr
    const int sel   = half * 2;       // +2 dwords for upper-half lanes (K split)

    // Per-lane index stream: lanes 0-15 read src[], lanes 16-31 read dst[]
    const int* ip = half ? dst : src;
    const int eMax = nEdges - 1;

    if (gwave >= nTiles) return;

    // Software pipeline: preload indices for the first tile (branch-free clamp).
    int idx = ip[min(gwave * 16 + l16, eMax)];

    for (int t = gwave; t < nTiles; t += nWaves) {
        const int base = t * 16;
        const int curIdx = idx;

        // Broadcast row indices first (one DScnt wait), then fire all gathers.
        uint32_t ridx[8];
        #pragma unroll
        for (int i = 0; i < 8; ++i)
            ridx[i] = (uint32_t)__shfl(curIdx, 4 * i + rsel);

        // 8 async coalesced gathers: instruction i pulls rows 4i..4i+3 (512B) into LDS.
        // GVS form: SGPR 64-bit feat base + 32-bit VGPR byte offset (max ~64MB, fits i32).
        #pragma unroll
        for (int i = 0; i < 8; ++i) {
            uint32_t voff  = ridx[i] * 128u + (uint32_t)(chunk * 16);
            uint32_t laddr = stBase + (uint32_t)((4 * i + rsel) * (ROW_DW * 4) + chunk * 16);
            asm volatile("global_load_async_to_lds_b128 %0, %1, %2"
                         :: "v"(laddr), "v"(voff), "s"(featBase) : "memory");
        }

        // Prefetch next tile's indices (branch-free; dummy on last iter is never used).
        // Latency hides under the async wait.
        idx = ip[min((t + nWaves) * 16 + l16, eMax)];

        asm volatile("s_wait_asynccnt 0" ::: "memory");

        // Chain 8 WMMAs over K = 32 in steps of 4.
        // A frag (16x4 f32): lanes 0-15 -> {K=k0,k0+1}, lanes 16-31 -> {K=k0+2,k0+3}, M = lane%16
        // B frag (4x16 f32): lanes 0-15 -> {K=k0,k0+1}, lanes 16-31 -> {K=k0+2,k0+3}, N = lane%16
        v8f c = {};
        #pragma unroll
        for (int k0 = 0; k0 < 32; k0 += 4) {
            v2f a = *(const v2f*)(st + l16 * ROW_DW        + k0 + sel);
            v2f b = *(const v2f*)(st + (16 + l16) * ROW_DW + k0 + sel);
            c = __builtin_amdgcn_wmma_f32_16x16x4_f32(
                    /*neg_a=*/false, a, /*neg_b=*/false, b,
                    /*c_mod=*/(short)0, c, /*reuse_a=*/false, /*reuse_b=*/false);
        }

        // Spill C to LDS, then lanes 0-15 pick up the diagonal C[m][m].
        {
            float* cl = cb + lane * C_DW;
            #pragma unroll
            for (int j = 0; j < 8; ++j) cl[j] = c[j];   // merges into 2x ds_store_b128
        }
        // C layout: VGPR r holds (M=r, N=lane) on lanes 0-15 and (M=r+8, N=lane-16) on 16-31.
        // Diagonal m: m<8 -> lane m, vgpr m ; m>=8 -> lane m+16, vgpr m-8.
        int m = l16;
        int srcLane = (m < 8) ? m : (m + 16);
        float s = cb[srcLane * C_DW + (m & 7)];
        if (lane < 16 && (base + m) < nEdges) out[base + m] = s;
    }
}

extern "C" void kernel_launch(void* const* d_in, const int* in_sizes, int n_in,
                              void* d_out, int out_size, void* d_ws, size_t ws_size,
                              hipStream_t stream) {
    const float* feat = (const float*)d_in[0];
    const int*   src  = (const int*)d_in[1];
    const int*   dst  = (const int*)d_in[2];
    float*       out  = (float*)d_out;

    int nEdges = in_sizes[1];
    int nTiles = (nEdges + 15) / 16;

    int blocks = 1024;
    if (blocks > nTiles) blocks = nTiles > 0 ? nTiles : 1;

    hipLaunchKernelGGL(hetero_dot_wmma_kernel, dim3(blocks), dim3(THREADS), 0, stream,
                       feat, src, dst, out, nEdges, nTiles);
}